// WaveletLayerND_84842783965587
// MI455X (gfx1250) — compile-verified
//
#include <hip/hip_runtime.h>

typedef __attribute__((ext_vector_type(2))) float v2f;
typedef __attribute__((ext_vector_type(8))) float v8f;

#define BATCH 4
#define OCH   32
#define ICH   32
#define HGT   128
#define WID   128
#define HW    (HGT * WID)
#define TH    32            // rows per spatial tile (2 rows per thread)
#define LDSW  (WID + 4)     // 132: padded to dodge bank conflicts
#define MH_CONST 0.8673250705840776f  // 2 / (sqrt(3) * pi^0.25)

// ---------------------------------------------------------------------------
// Kernel 1: fused (x - t)/s -> mexican-hat psi -> grouped 3x3 conv.
// One workgroup per (b, o, 32-row tile). Loop over i: build psi tile (zero
// halo, since the reference zero-pads the *wavelet* tensor) in LDS, then each
// thread computes a 2x8 output patch from a 4x10 register window (2.5 LDS
// dwords per pixel). The next channel's x tile is prefetched during the conv.
// ---------------------------------------------------------------------------
__global__ __launch_bounds__(256) void wavelet_gconv_kernel(
    const float* __restrict__ x,      // (B, I, H, W)
    const float* __restrict__ scale,  // (O, I)
    const float* __restrict__ trans,  // (O, I)
    const float* __restrict__ wconv,  // (O, I, 3, 3)
    float* __restrict__ y)            // (B, O, H, W)
{
    __shared__ float psi[(TH + 2) * LDSW];   // 34 x 132 floats = 17.6 KB

    const int tid  = threadIdx.x;
    const int idx  = blockIdx.x;
    const int tile = idx & 3;            // 4 row-tiles of 32 rows
    const int o    = (idx >> 2) & 31;
    const int b    = idx >> 7;
    const int row0 = tile * TH;

    const int rowPair = (tid >> 4) * 2;  // 0,2,...,30
    const int colBase = (tid & 15) * 8;  // 8 consecutive pixels (32B aligned)

    float acc[2][8];
#pragma unroll
    for (int rr = 0; rr < 2; ++rr)
#pragma unroll
        for (int j = 0; j < 8; ++j) acc[rr][j] = 0.0f;

    for (int i = 0; i < ICH; ++i) {
        const int oi    = o * ICH + i;
        const float t   = trans[oi];
        const float inv = 1.0f / scale[oi];

        float wk[9];
#pragma unroll
        for (int k = 0; k < 9; ++k) wk[k] = wconv[oi * 9 + k];

        const float* xin = x + (size_t)(b * ICH + i) * HW;

        // Fill (TH+2) x (WID+2) psi tile; halo outside the image is 0.
        for (int e = tid; e < (TH + 2) * (WID + 2); e += 256) {
            const int r  = e / (WID + 2);
            const int c  = e - r * (WID + 2);
            const int gh = row0 + r - 1;
            const int gw = c - 1;
            float p = 0.0f;
            if (gh >= 0 && gh < HGT && gw >= 0 && gw < WID) {
                const float u  = (xin[gh * WID + gw] - t) * inv;
                const float u2 = u * u;
                p = MH_CONST * (u2 - 1.0f) * __expf(-0.5f * u2);
            }
            psi[r * LDSW + c] = p;
        }
        __syncthreads();

        // Prefetch next channel's tile (34 rows x 4 cachelines of 128B)
        // so HBM/L2 latency overlaps with the conv FMAs below.
        if (i + 1 < ICH && tid < 136) {
            const int r   = tid >> 2;          // 0..33
            const int seg = (tid & 3) * 32;    // 0,32,64,96 floats
            const int gh  = row0 + r - 1;
            if (gh >= 0 && gh < HGT) {
                const float* nxt = x + (size_t)(b * ICH + i + 1) * HW;
                __builtin_prefetch(nxt + gh * WID + seg, 0, 3);
            }
        }

        // Load 4x10 window into registers, compute 2x8 patch (144 FMA).
        float line[4][10];
#pragma unroll
        for (int rr = 0; rr < 4; ++rr)
#pragma unroll
            for (int cc = 0; cc < 10; ++cc)
                line[rr][cc] = psi[(rowPair + rr) * LDSW + colBase + cc];

#pragma unroll
        for (int rr = 0; rr < 2; ++rr)
#pragma unroll
            for (int j = 0; j < 8; ++j) {
                float s = 0.0f;
#pragma unroll
                for (int kh = 0; kh < 3; ++kh)
#pragma unroll
                    for (int kw = 0; kw < 3; ++kw)
                        s += wk[kh * 3 + kw] * line[rr + kh][j + kw];
                acc[rr][j] += s;
            }
        __syncthreads();
    }

#pragma unroll
    for (int rr = 0; rr < 2; ++rr) {
        float* yout = y + (size_t)(b * OCH + o) * HW
                        + (row0 + rowPair + rr) * WID + colBase;
#pragma unroll
        for (int j = 0; j < 8; ++j) yout[j] = acc[rr][j];
    }
}

// ---------------------------------------------------------------------------
// Kernel 2: final 1x1 conv = per-batch GEMM out(32 x 16384) = Wf(32x32) * y.
// One wave per 16x16 output tile, K=32 in 8 steps of V_WMMA_F32_16X16X4_F32.
// VGPR layouts per CDNA5 ISA 7.12.2:
//   A 16x4 : lanes 0-15 -> K=0,1 in v0,v1 ; lanes 16-31 -> K=2,3
//   B 4x16 : lanes 0-15 -> rows K=0,1     ; lanes 16-31 -> rows K=2,3
//   D 16x16: vgpr j -> M=j (lanes 0-15) / M=j+8 (lanes 16-31), N=lane&15
// ---------------------------------------------------------------------------
__global__ __launch_bounds__(256) void mix1x1_wmma_kernel(
    const float* __restrict__ y,   // (B, O, H*W)
    const float* __restrict__ wf,  // (O2, O)
    float* __restrict__ out)       // (B, O2, H*W)
{
    const int lane   = threadIdx.x & 31;
    const int waveId = blockIdx.x * (blockDim.x >> 5) + (threadIdx.x >> 5);

    // 8192 waves = b(4) x mtile(2) x ntile(1024)
    const int b     = waveId >> 11;
    const int mtile = (waveId >> 10) & 1;
    const int ntile = waveId & 1023;
    const int m0    = mtile * 16;
    const int n0    = ntile * 16;

    const int half = lane >> 4;   // 0 or 1
    const int l15  = lane & 15;
    const int ncol = n0 + l15;

    const float* yb = y + (size_t)b * OCH * HW;

    v8f c = {};
#pragma unroll
    for (int k0 = 0; k0 < OCH; k0 += 4) {
        const int kb = k0 + half * 2;
        v2f a, bb;
        a.x  = wf[(m0 + l15) * OCH + kb];
        a.y  = wf[(m0 + l15) * OCH + kb + 1];
        bb.x = yb[(size_t)kb * HW + ncol];
        bb.y = yb[(size_t)(kb + 1) * HW + ncol];
        c = __builtin_amdgcn_wmma_f32_16x16x4_f32(
                /*neg_a=*/false, a, /*neg_b=*/false, bb,
                /*c_mod=*/(short)0, c, /*reuse_a=*/false, /*reuse_b=*/false);
    }

    float* ob = out + (size_t)b * OCH * HW;
#pragma unroll
    for (int j = 0; j < 8; ++j) {
        const int m = m0 + half * 8 + j;
        ob[(size_t)m * HW + ncol] = c[j];
    }
}

// ---------------------------------------------------------------------------
extern "C" void kernel_launch(void* const* d_in, const int* in_sizes, int n_in,
                              void* d_out, int out_size, void* d_ws, size_t ws_size,
                              hipStream_t stream) {
    (void)in_sizes; (void)n_in; (void)out_size; (void)ws_size;

    const float* x  = (const float*)d_in[0];  // (4,32,128,128)
    const float* sc = (const float*)d_in[1];  // (1,32,32,1,1)
    const float* tr = (const float*)d_in[2];  // (1,32,32,1,1)
    const float* wc = (const float*)d_in[3];  // (32,32,3,3)
    const float* wf = (const float*)d_in[4];  // (32,32,1,1)

    float* y   = (float*)d_ws;   // 4*32*128*128*4 = 8.39 MB scratch
    float* out = (float*)d_out;

    // Kernel 1: B*O*(H/TH) = 4*32*4 = 512 workgroups of 256 threads (8 waves)
    hipLaunchKernelGGL(wavelet_gconv_kernel, dim3(BATCH * OCH * (HGT / TH)),
                       dim3(256), 0, stream, x, sc, tr, wc, y);

    // Kernel 2: 8192 waves / 8 waves-per-block = 1024 blocks
    hipLaunchKernelGGL(mix1x1_wmma_kernel, dim3(1024), dim3(256), 0, stream,
                       y, wf, out);
}